// GCNIIBackbone_57166014710044
// MI455X (gfx1250) — compile-verified
//
#include <hip/hip_runtime.h>
#include <math.h>

#define N_NODES 10000
#define N_EDGES 320000
#define DIN 128
#define DMID 256
#define DOUT 128
#define N_LAYERS 8

typedef __attribute__((ext_vector_type(2))) float v2f;
typedef __attribute__((ext_vector_type(8))) float v8f;

__device__ __forceinline__ v8f wmma_f32_k4(v2f a, v2f b, v8f c) {
  // D = A(16x4,f32) * B(4x16,f32) + C(16x16,f32)
  return __builtin_amdgcn_wmma_f32_16x16x4_f32(false, a, false, b, (short)0, c,
                                               false, false);
}

__device__ __forceinline__ void atomAddF(float* p, float v) {
  unsafeAtomicAdd(p, v);  // global_atomic_add_f32 on gfx1250
}

// ---------------------------------------------------------------------------
// GCN norm setup
// ---------------------------------------------------------------------------
__global__ void deg_init_kernel(float* __restrict__ deg) {
  int v = blockIdx.x * 256 + threadIdx.x;
  if (v < N_NODES) deg[v] = 1.0f;  // self loop
}

__global__ void deg_accum_kernel(const int* __restrict__ ei,
                                 float* __restrict__ deg) {
  int e = blockIdx.x * 256 + threadIdx.x;
  if (e < N_EDGES) atomAddF(&deg[ei[N_EDGES + e]], 1.0f);
}

__global__ void deg_rsqrt_kernel(float* __restrict__ deg) {
  int v = blockIdx.x * 256 + threadIdx.x;
  if (v < N_NODES) deg[v] = rsqrtf(deg[v]);  // deg >= 1 always
}

// agg[v][:] = dinv[v]^2 * f[v][:]   (self-loop term, also zero-inits agg)
__global__ void selfloop_init_kernel(const float* __restrict__ f,
                                     const float* __restrict__ dinv,
                                     float* __restrict__ agg) {
  int idx = blockIdx.x * 256 + threadIdx.x;
  const int tot = N_NODES * (DMID / 4);
  if (idx < tot) {
    int v = idx / (DMID / 4);
    float d = dinv[v];
    float s = d * d;
    float4 val = ((const float4*)f)[idx];
    float4 o;
    o.x = s * val.x; o.y = s * val.y; o.z = s * val.z; o.w = s * val.w;
    ((float4*)agg)[idx] = o;
  }
}

// One wave per edge: agg[col] += dinv[row]*dinv[col] * f[row]
__global__ __launch_bounds__(256) void edge_scatter_kernel(
    const int* __restrict__ ei, const float* __restrict__ dinv,
    const float* __restrict__ f, float* __restrict__ agg) {
  int e = blockIdx.x * 8 + (threadIdx.x >> 5);
  if (e >= N_EDGES) return;
  int lane = threadIdx.x & 31;
  int r = ei[e];
  int c = ei[N_EDGES + e];
  float nrm = dinv[r] * dinv[c];
  const float4* src = (const float4*)(f + (size_t)r * DMID);
  float* dst = agg + (size_t)c * DMID;
  float4 v0 = src[2 * lane];
  float4 v1 = src[2 * lane + 1];
  int b = lane * 8;
  atomAddF(dst + b + 0, nrm * v0.x);
  atomAddF(dst + b + 1, nrm * v0.y);
  atomAddF(dst + b + 2, nrm * v0.z);
  atomAddF(dst + b + 3, nrm * v0.w);
  atomAddF(dst + b + 4, nrm * v1.x);
  atomAddF(dst + b + 5, nrm * v1.y);
  atomAddF(dst + b + 6, nrm * v1.z);
  atomAddF(dst + b + 7, nrm * v1.w);
}

// ---------------------------------------------------------------------------
// Weight pre-pack into WMMA B-fragment order:
//   Wp[((n16 * K/4 + ks) * 32 + lane)] = { W[4ks+2h][n16*16+l15],
//                                          W[4ks+2h+1][n16*16+l15] }
// -> in the GEMM loop each lane does ONE contiguous b64 load per WMMA,
//    with compile-time immediate offsets (no address VALU).
// ---------------------------------------------------------------------------
__global__ void wpack_kernel(const float* __restrict__ W, v2f* __restrict__ Wp,
                             int K, int N) {
  int idx = blockIdx.x * 256 + threadIdx.x;
  int total = (N / 16) * (K / 4) * 32;
  if (idx >= total) return;
  int lane = idx & 31;
  int t = idx >> 5;
  int ks = t % (K / 4);
  int n16 = t / (K / 4);
  int h = lane >> 4, l15 = lane & 15;
  int k = 4 * ks + 2 * h;
  int n = n16 * 16 + l15;
  v2f v;
  v.x = W[(size_t)k * N + n];
  v.y = W[(size_t)(k + 1) * N + n];
  Wp[idx] = v;
}

// ---------------------------------------------------------------------------
// Dense GEMM (packed B): C = relu(A[M,K] @ W[K,N] + bias[N])
// 8 waves/block, wave -> 16x32 tile, block -> 32 x 128
// ---------------------------------------------------------------------------
template <int K>
__global__ __launch_bounds__(256) void gemm_bias_relu_kernel(
    const float* __restrict__ A, const v2f* __restrict__ Wp,
    const float* __restrict__ bias, float* __restrict__ C, int M, int N) {
  constexpr int KS = K / 4;
  const int lane = threadIdx.x & 31;
  const int wave = threadIdx.x >> 5;
  const int mb = blockIdx.x * 32 + (wave >> 2) * 16;
  const int nb = blockIdx.y * 128 + (wave & 3) * 32;
  const int h = lane >> 4;
  const int l15 = lane & 15;

  int mrow = mb + l15;
  if (mrow > M - 1) mrow = M - 1;  // clamp: loads in-bounds, EXEC stays all-1
  const v2f* ap = (const v2f*)(A + (size_t)mrow * K + 2 * h);
  const v2f* bp0 = Wp + (size_t)(nb >> 4) * KS * 32 + lane;
  const v2f* bp1 = bp0 + KS * 32;

  v8f acc0 = {}, acc1 = {};
#pragma unroll
  for (int ks = 0; ks < KS; ++ks) {
    v2f a = ap[ks * 2];
    acc0 = wmma_f32_k4(a, bp0[ks * 32], acc0);
    acc1 = wmma_f32_k4(a, bp1[ks * 32], acc1);
  }

  float bn0 = bias[nb + l15];
  float bn1 = bias[nb + 16 + l15];
#pragma unroll
  for (int r = 0; r < 8; ++r) {
    int m = mb + r + 8 * h;
    if (m < M) {
      C[(size_t)m * N + nb + l15] = fmaxf(acc0[r] + bn0, 0.0f);
      C[(size_t)m * N + nb + 16 + l15] = fmaxf(acc1[r] + bn1, 0.0f);
    }
  }
}

// ---------------------------------------------------------------------------
// Fused GCNII layer (ALPHA == 0.5 -> both GEMMs share coefficient beta/2):
//   acc  = agg @ w1 + f0 @ w2       (two K=256 passes into one accumulator)
//   fout = relu(c0*(agg + f0) + c1*acc), c0 = (1-beta)/2, c1 = beta/2
// 8 waves/block, wave -> 16x64 tile (4 accumulators), block -> 32 x 256
// ---------------------------------------------------------------------------
__global__ __launch_bounds__(256) void layer_fused_kernel(
    const float* __restrict__ agg, const float* __restrict__ f0,
    const v2f* __restrict__ w1p, const v2f* __restrict__ w2p,
    float* __restrict__ fout, float c0f, float c1f) {
  constexpr int M = N_NODES, K = DMID, N = DMID, KS = K / 4;
  const int lane = threadIdx.x & 31;
  const int wave = threadIdx.x >> 5;
  const int mb = blockIdx.x * 32 + (wave >> 2) * 16;
  const int nb = (wave & 3) * 64;
  const int h = lane >> 4;
  const int l15 = lane & 15;

  int mrow = mb + l15;
  if (mrow > M - 1) mrow = M - 1;
  const v2f* ap1 = (const v2f*)(agg + (size_t)mrow * K + 2 * h);
  const v2f* ap2 = (const v2f*)(f0 + (size_t)mrow * K + 2 * h);
  const v2f* b1 = w1p + (size_t)(nb >> 4) * KS * 32 + lane;
  const v2f* b2 = w2p + (size_t)(nb >> 4) * KS * 32 + lane;

  v8f acc0 = {}, acc1 = {}, acc2 = {}, acc3 = {};
#pragma unroll
  for (int ks = 0; ks < KS; ++ks) {
    v2f a = ap1[ks * 2];
    acc0 = wmma_f32_k4(a, b1[ks * 32], acc0);
    acc1 = wmma_f32_k4(a, b1[KS * 32 + ks * 32], acc1);
    acc2 = wmma_f32_k4(a, b1[2 * KS * 32 + ks * 32], acc2);
    acc3 = wmma_f32_k4(a, b1[3 * KS * 32 + ks * 32], acc3);
  }
#pragma unroll
  for (int ks = 0; ks < KS; ++ks) {
    v2f a = ap2[ks * 2];
    acc0 = wmma_f32_k4(a, b2[ks * 32], acc0);
    acc1 = wmma_f32_k4(a, b2[KS * 32 + ks * 32], acc1);
    acc2 = wmma_f32_k4(a, b2[2 * KS * 32 + ks * 32], acc2);
    acc3 = wmma_f32_k4(a, b2[3 * KS * 32 + ks * 32], acc3);
  }

#pragma unroll
  for (int r = 0; r < 8; ++r) {
    int m = mb + r + 8 * h;
    if (m < M) {
      size_t base = (size_t)m * N + nb + l15;
      float s0 = agg[base] + f0[base];
      float s1 = agg[base + 16] + f0[base + 16];
      float s2 = agg[base + 32] + f0[base + 32];
      float s3 = agg[base + 48] + f0[base + 48];
      fout[base] = fmaxf(c0f * s0 + c1f * acc0[r], 0.0f);
      fout[base + 16] = fmaxf(c0f * s1 + c1f * acc1[r], 0.0f);
      fout[base + 32] = fmaxf(c0f * s2 + c1f * acc2[r], 0.0f);
      fout[base + 48] = fmaxf(c0f * s3 + c1f * acc3[r], 0.0f);
    }
  }
}

// ---------------------------------------------------------------------------
// Host launcher
// ---------------------------------------------------------------------------
extern "C" void kernel_launch(void* const* d_in, const int* in_sizes, int n_in,
                              void* d_out, int out_size, void* d_ws,
                              size_t ws_size, hipStream_t stream) {
  const float* x = (const float*)d_in[0];
  const int* ei = (const int*)d_in[1];  // [2, N_EDGES]
  const float* lin1_w = (const float*)d_in[2];
  const float* lin1_b = (const float*)d_in[3];
  const float* w1 = (const float*)d_in[4];  // [8,256,256]
  const float* w2 = (const float*)d_in[5];
  const float* lin2_w = (const float*)d_in[6];
  const float* lin2_b = (const float*)d_in[7];
  float* out = (float*)d_out;

  // workspace carve-out
  char* ws = (char*)d_ws;
  float* deg = (float*)ws;
  ws += ((N_NODES * 4 + 255) / 256) * 256;
  float* f0 = (float*)ws;
  ws += (size_t)N_NODES * DMID * 4;
  float* f = (float*)ws;
  ws += (size_t)N_NODES * DMID * 4;
  float* agg = (float*)ws;
  ws += (size_t)N_NODES * DMID * 4;
  // packed weights (v2f each): lin1, 8x w1, 8x w2, lin2
  const int PK1 = (DMID / 16) * (DIN / 4) * 32;   // lin1: 16384
  const int PKL = (DMID / 16) * (DMID / 4) * 32;  // per-layer: 32768
  const int PK2 = (DOUT / 16) * (DMID / 4) * 32;  // lin2: 16384
  v2f* lin1_p = (v2f*)ws;
  ws += (size_t)PK1 * sizeof(v2f);
  v2f* w1_p = (v2f*)ws;
  ws += (size_t)N_LAYERS * PKL * sizeof(v2f);
  v2f* w2_p = (v2f*)ws;
  ws += (size_t)N_LAYERS * PKL * sizeof(v2f);
  v2f* lin2_p = (v2f*)ws;

  // pre-pack all weights into WMMA fragment order
  wpack_kernel<<<(PK1 + 255) / 256, 256, 0, stream>>>(lin1_w, lin1_p, DIN,
                                                      DMID);
  for (int i = 0; i < N_LAYERS; ++i) {
    wpack_kernel<<<(PKL + 255) / 256, 256, 0, stream>>>(
        w1 + (size_t)i * DMID * DMID, w1_p + (size_t)i * PKL, DMID, DMID);
    wpack_kernel<<<(PKL + 255) / 256, 256, 0, stream>>>(
        w2 + (size_t)i * DMID * DMID, w2_p + (size_t)i * PKL, DMID, DMID);
  }
  wpack_kernel<<<(PK2 + 255) / 256, 256, 0, stream>>>(lin2_w, lin2_p, DMID,
                                                      DOUT);

  // GCN normalization: deg -> dinv
  deg_init_kernel<<<(N_NODES + 255) / 256, 256, 0, stream>>>(deg);
  deg_accum_kernel<<<(N_EDGES + 255) / 256, 256, 0, stream>>>(ei, deg);
  deg_rsqrt_kernel<<<(N_NODES + 255) / 256, 256, 0, stream>>>(deg);

  const int gm = (N_NODES + 31) / 32;  // 313

  // f0 = relu(x @ lin1_w + lin1_b)
  gemm_bias_relu_kernel<DIN><<<dim3(gm, DMID / 128), 256, 0, stream>>>(
      x, lin1_p, lin1_b, f0, N_NODES, DMID);

  const float* fcur = f0;
  for (int i = 0; i < N_LAYERS; ++i) {
    selfloop_init_kernel<<<(N_NODES * (DMID / 4) + 255) / 256, 256, 0,
                           stream>>>(fcur, deg, agg);
    edge_scatter_kernel<<<(N_EDGES + 7) / 8, 256, 0, stream>>>(ei, deg, fcur,
                                                               agg);
    float beta = logf(1.0f / (float)(i + 1) + 1.0f);
    float c0 = 0.5f * (1.0f - beta);
    float c1 = 0.5f * beta;
    layer_fused_kernel<<<dim3(gm, 1), 256, 0, stream>>>(
        agg, f0, w1_p + (size_t)i * PKL, w2_p + (size_t)i * PKL, f, c0, c1);
    fcur = f;
  }

  // out = relu(f @ lin2_w + lin2_b)
  gemm_bias_relu_kernel<DMID><<<dim3(gm, DOUT / 128), 256, 0, stream>>>(
      fcur, lin2_p, lin2_b, out, N_NODES, DOUT);
}